// SimpleGraphConvolution_90022514524541
// MI455X (gfx1250) — compile-verified
//
#include <hip/hip_runtime.h>
#include <hip/hip_bf16.h>

typedef float v2f __attribute__((ext_vector_type(2)));
typedef float v8f __attribute__((ext_vector_type(8)));

#define N_NODES 100000
#define N_EDGES 1600000
#define IN_F    256
#define OUT_F   64
#define NTILES  (N_NODES / 16)   // 6250 exactly

// ---------------------------------------------------------------------------
// Hardware f32 scatter-add at device scope (L2 atomic units). Guarantees
// global_atomic_add_f32 (no CAS loop) for the 307M atomic adds across hops.
// ---------------------------------------------------------------------------
__device__ __forceinline__ void atomic_add_f32_dev(float* p, float v) {
    asm volatile("global_atomic_add_f32 %0, %1, off scope:SCOPE_DEV"
                 :: "v"(p), "v"(v) : "memory");
}

// ---------------------------------------------------------------------------
// Dense projection: out[n, :] = x[n, :] @ W.T + b   using V_WMMA_F32_16X16X4_F32
// One wave computes a 16(node) x 64(feature) tile: 4 accumulators of 16x16.
// A (16x4 f32): lanes 0-15 hold M=lane, VGPR0=K0, VGPR1=K1; lanes 16-31 K2/K3.
//   -> per-lane float2 load at x[m*256 + k + 2*(lane>>4)]
// B (4x16 f32) mirrors A with n = lane&15 indexing rows of W (cols of W.T).
// C/D (16x16 f32): VGPR r -> row r (lanes 0-15) / row r+8 (lanes 16-31), col lane&15.
// ---------------------------------------------------------------------------
__global__ void sgc_gemm_wmma(const float* __restrict__ x,
                              const float* __restrict__ W,
                              const float* __restrict__ bias,
                              float* __restrict__ out) {
    const int wave = (blockIdx.x * blockDim.x + threadIdx.x) >> 5;
    const int lane = threadIdx.x & 31;
    if (wave >= NTILES) return;               // whole wave exits -> EXEC stays all-1s

    const int m    = lane & 15;               // row within tile (A) / col (B,C)
    const int koff = (lane >> 4) << 1;        // 0 for lanes 0-15, 2 for lanes 16-31

    const float* __restrict__ xrow = x + (size_t)(wave * 16 + m) * IN_F;
    const float* __restrict__ wr0  = W + (size_t)(m +  0) * IN_F;  // n-tile 0
    const float* __restrict__ wr1  = W + (size_t)(m + 16) * IN_F;  // n-tile 1
    const float* __restrict__ wr2  = W + (size_t)(m + 32) * IN_F;  // n-tile 2
    const float* __restrict__ wr3  = W + (size_t)(m + 48) * IN_F;  // n-tile 3

    v8f acc0 = {}, acc1 = {}, acc2 = {}, acc3 = {};

#pragma unroll 4
    for (int k = 0; k < IN_F; k += 4) {
        v2f a  = *(const v2f*)(xrow + k + koff);
        v2f b0 = *(const v2f*)(wr0  + k + koff);
        v2f b1 = *(const v2f*)(wr1  + k + koff);
        v2f b2 = *(const v2f*)(wr2  + k + koff);
        v2f b3 = *(const v2f*)(wr3  + k + koff);
        acc0 = __builtin_amdgcn_wmma_f32_16x16x4_f32(false, a, false, b0, (short)0, acc0, false, false);
        acc1 = __builtin_amdgcn_wmma_f32_16x16x4_f32(false, a, false, b1, (short)0, acc1, false, false);
        acc2 = __builtin_amdgcn_wmma_f32_16x16x4_f32(false, a, false, b2, (short)0, acc2, false, false);
        acc3 = __builtin_amdgcn_wmma_f32_16x16x4_f32(false, a, false, b3, (short)0, acc3, false, false);
    }

    const int row_base = wave * 16 + ((lane >> 4) ? 8 : 0);
    const int col      = lane & 15;
    const float bv0 = bias[col +  0];
    const float bv1 = bias[col + 16];
    const float bv2 = bias[col + 32];
    const float bv3 = bias[col + 48];

#pragma unroll
    for (int r = 0; r < 8; ++r) {
        float* o = out + (size_t)(row_base + r) * OUT_F + col;
        o[ 0] = acc0[r] + bv0;
        o[16] = acc1[r] + bv1;
        o[32] = acc2[r] + bv2;
        o[48] = acc3[r] + bv3;
    }
}

// ---------------------------------------------------------------------------
// Zero-fill (float4 grid-stride).
// ---------------------------------------------------------------------------
__global__ void sgc_zero(float4* __restrict__ p, int n4) {
    const float4 z = {0.f, 0.f, 0.f, 0.f};
    for (int i = blockIdx.x * blockDim.x + threadIdx.x; i < n4;
         i += gridDim.x * blockDim.x)
        p[i] = z;
}

// ---------------------------------------------------------------------------
// One SpMM hop: nxt[dst,:] += w_e * cur[src,:]  (COO scatter).
// One wave per edge; lane handles float2 at feature 2*lane (wave covers the
// full 256B row contiguously -> single global_load_b64 gather per lane).
// cur/nxt are 25.6 MB each -> resident in the 192 MB L2; edge lists are
// streamed with TH_NT so they don't evict the feature working set.
// ---------------------------------------------------------------------------
__global__ void sgc_scatter(const float* __restrict__ cur,
                            float* __restrict__ nxt,
                            const int* __restrict__ esrc,
                            const int* __restrict__ edst,
                            const float* __restrict__ ew,
                            int n_edges) {
    const int gid  = blockIdx.x * blockDim.x + threadIdx.x;
    const int e    = gid >> 5;
    const int lane = gid & 31;
    if (e >= n_edges) return;

    const int   s = __builtin_nontemporal_load(esrc + e);
    const int   d = __builtin_nontemporal_load(edst + e);
    const float w = __builtin_nontemporal_load(ew + e);

    const float* __restrict__ ps = cur + (size_t)s * OUT_F + lane * 2;
    float*       __restrict__ pd = nxt + (size_t)d * OUT_F + lane * 2;

    const v2f vv = *(const v2f*)ps;
    atomic_add_f32_dev(pd + 0, vv.x * w);
    atomic_add_f32_dev(pd + 1, vv.y * w);
}

// ---------------------------------------------------------------------------
extern "C" void kernel_launch(void* const* d_in, const int* in_sizes, int n_in,
                              void* d_out, int out_size, void* d_ws, size_t ws_size,
                              hipStream_t stream) {
    const float* x    = (const float*)d_in[0];
    const float* W    = (const float*)d_in[1];
    const float* b    = (const float*)d_in[2];
    const int*   esrc = (const int*)d_in[3];
    const int*   edst = (const int*)d_in[4];
    const float* ew   = (const float*)d_in[5];

    float* out = (float*)d_out;                 // N_NODES * OUT_F floats
    float* ws  = (float*)d_ws;                  // ping-pong buffer (25.6 MB)

    const int n4 = N_NODES * OUT_F / 4;         // float4 count for zero fill

    // GEMM -> ws  (projection, WMMA f32)
    sgc_gemm_wmma<<<(NTILES + 7) / 8, 256, 0, stream>>>(x, W, b, ws);

    const dim3 sgrid((unsigned)(((long long)N_EDGES * 32 + 255) / 256));

    // hop 1: out = A @ ws
    sgc_zero<<<1024, 256, 0, stream>>>((float4*)out, n4);
    sgc_scatter<<<sgrid, 256, 0, stream>>>(ws, out, esrc, edst, ew, N_EDGES);

    // hop 2: ws = A @ out
    sgc_zero<<<1024, 256, 0, stream>>>((float4*)ws, n4);
    sgc_scatter<<<sgrid, 256, 0, stream>>>(out, ws, esrc, edst, ew, N_EDGES);

    // hop 3: out = A @ ws   (final result lands in d_out)
    sgc_zero<<<1024, 256, 0, stream>>>((float4*)out, n4);
    sgc_scatter<<<sgrid, 256, 0, stream>>>(ws, out, esrc, edst, ew, N_EDGES);
}